// TotalDegree_29755533426739
// MI455X (gfx1250) — compile-verified
//
#include <hip/hip_runtime.h>
#include <stdint.h>

// TotalDegree polynomial features: x[B,8] -> out[B,495] (all monomials, deg<=4)
// Store-bandwidth-bound: 130MB out @ 23.3TB/s => ~5.7us floor. Strategy:
//  - per-thread fully-unrolled DFS product tree (emission order == Python sorted order)
//  - stage block's 64 rows x 495 terms contiguously in LDS (126,720B)
//  - one TDM tensor_store_from_lds per block (contiguous 1-D tile) + s_wait_tensorcnt

#define DIMV 8
#define NT   495
#define ROWS 64
#define TPB  64

typedef uint32_t v4u __attribute__((ext_vector_type(4)));
typedef int      v8i __attribute__((ext_vector_type(8)));
typedef int      v4i __attribute__((ext_vector_type(4)));

__global__ __launch_bounds__(TPB)
void TotalDegree_29755533426739_kernel(const float* __restrict__ x,
                                       float* __restrict__ out) {
  __shared__ float smem[ROWS * NT];       // 126,720 B: block's output tile, contiguous
  const int t = threadIdx.x;
  const uint32_t row = blockIdx.x * ROWS + t;

  // Load this row's 8 floats (row = 32B, 16B-aligned -> two float4 loads).
  const float4* xin = (const float4*)x;
  const float4 lo = xin[2u * row];
  const float4 hi = xin[2u * row + 1u];
  const float xv[DIMV] = {lo.x, lo.y, lo.z, lo.w, hi.x, hi.y, hi.z, hi.w};

  // Pre-order DFS over non-decreasing index tuples of length 0..4:
  // emission order exactly matches sorted(combinations_with_replacement) chain.
  float* rp = smem + t * NT;
  int k = 0;
  rp[k++] = 1.0f;                                   // ()
#pragma unroll
  for (int i = 0; i < DIMV; ++i) {
    const float vi = xv[i];
    rp[k++] = vi;                                   // (i)
#pragma unroll
    for (int j = i; j < DIMV; ++j) {
      const float vj = vi * xv[j];
      rp[k++] = vj;                                 // (i,j)
#pragma unroll
      for (int l = j; l < DIMV; ++l) {
        const float vl = vj * xv[l];
        rp[k++] = vl;                               // (i,j,l)
#pragma unroll
        for (int m = l; m < DIMV; ++m) {
          rp[k++] = vl * xv[m];                     // (i,j,l,m)
        }
      }
    }
  }
  __syncthreads();

#if __has_builtin(__builtin_amdgcn_tensor_store_from_lds)
  // ---- TDM path: one DMA descriptor moves the whole 126,720B tile LDS->global.
  if (t == 0) {
    const uint64_t gaddr =
        (uint64_t)(uintptr_t)out + (uint64_t)blockIdx.x * (uint64_t)(ROWS * NT * 4);
    const uint32_t laddr = (uint32_t)(uintptr_t)smem;  // low 32 bits = LDS byte offset
    const uint32_t nelem = (uint32_t)(ROWS * NT);      // 31680 elements of 4B

    // D# group 0 (128b): count=1 | lds_addr | global_addr[120:64] | type=2
    v4u g0;
    g0[0] = 1u;                                        // count=1 (valid descriptor)
    g0[1] = laddr;                                     // lds_addr (bytes)
    g0[2] = (uint32_t)gaddr;                           // global_addr[31:0] -> bits 95:64
    g0[3] = (uint32_t)((gaddr >> 32) & 0x01FFFFFFu)    // global_addr[56:32]
            | 0x80000000u;                             // type = 2 ("image")
    // D# group 1 (256b)
    v8i g1;
    g1[0] = (int)(2u << 16);                           // data_size=2 (4B); mask=0
    g1[1] = (int)((nelem & 0xFFFFu) << 16);            // tensor_dim0[15:0]
    g1[2] = (int)(((nelem >> 16) & 0xFFFFu) | (1u << 16)); // dim0 hi | tensor_dim1=1
    g1[3] = (int)((nelem & 0xFFFFu) << 16);            // tile_dim0 = 31680
    g1[4] = 0;                                         // tile_dim1=0, tile_dim2=0 (unused)
    g1[5] = (int)nelem;                                // tensor_dim0_stride[31:0]
    g1[6] = 0;                                         // stride0 hi | stride1 lo
    g1[7] = 0;
    v4i g2 = {0, 0, 0, 0};
    v4i g3 = {0, 0, 0, 0};
#if defined(__clang_major__) && (__clang_major__ >= 23)
    v8i g4 = {0, 0, 0, 0, 0, 0, 0, 0};
    __builtin_amdgcn_tensor_store_from_lds(g0, g1, g2, g3, g4, 0);
#else
    __builtin_amdgcn_tensor_store_from_lds(g0, g1, g2, g3, 0);
#endif
    __builtin_amdgcn_s_wait_tensorcnt(0);
  }
#else
  // ---- Fallback: async LDS->global b128 stores (ASYNCcnt path), fully coalesced.
  const float4* ls4 = (const float4*)smem;
  float4* g4p = (float4*)(out + (size_t)blockIdx.x * (ROWS * NT));
  const uint32_t lbase = (uint32_t)(uintptr_t)ls4;
  constexpr int NQ = ROWS * NT / 4;          // 7920 float4 per block
  constexpr int FULL = NQ / TPB;             // 123 full rounds
#pragma unroll 1
  for (int it = 0; it < FULL; ++it) {
    const int idx = it * TPB + t;
    const uint64_t ga = (uint64_t)(uintptr_t)(g4p + idx);
    const uint32_t la = lbase + 16u * (uint32_t)idx;
    asm volatile("global_store_async_from_lds_b128 %0, %1, off"
                 :: "v"(ga), "v"(la) : "memory");
  }
  if (t < NQ - FULL * TPB) {                 // 48-float4 tail via normal store
    const int idx = FULL * TPB + t;
    g4p[idx] = ls4[idx];
  }
  asm volatile("s_wait_asynccnt 0" ::: "memory");
#endif
}

extern "C" void kernel_launch(void* const* d_in, const int* in_sizes, int n_in,
                              void* d_out, int out_size, void* d_ws, size_t ws_size,
                              hipStream_t stream) {
  (void)n_in; (void)out_size; (void)d_ws; (void)ws_size;
  const float* x = (const float*)d_in[0];
  float* out = (float*)d_out;
  const int rows = in_sizes[0] / DIMV;       // 65536
  const int blocks = rows / ROWS;            // 1024
  TotalDegree_29755533426739_kernel<<<blocks, TPB, 0, stream>>>(x, out);
}